// ChamferDistance_59631325938052
// MI455X (gfx1250) — compile-verified
//
#include <hip/hip_runtime.h>

typedef __attribute__((ext_vector_type(2))) float v2f;
typedef __attribute__((ext_vector_type(8))) float v8f;

#define CHAMFER_THRESH 33.33f

__global__ void chamfer_zero_acc(float* acc, int n) {
    int i = blockIdx.x * blockDim.x + threadIdx.x;
    if (i < n) acc[i] = 0.0f;
}

// One wave handles 32 source rows (two 16-row A tiles); 8 waves/block -> 256 rows.
// blockIdx.x: row tile, blockIdx.y: batch, blockIdx.z: direction (0: S->T, 1: T->S).
__global__ __launch_bounds__(256) void chamfer_dir_kernel(
    const float* __restrict__ pcA, const float* __restrict__ pcB,
    float* __restrict__ acc, int npts) {

    const int batch = blockIdx.y;
    const int dir   = blockIdx.z;
    const float* S = (dir == 0 ? pcA : pcB) + (size_t)batch * npts * 3;
    const float* T = (dir == 0 ? pcB : pcA) + (size_t)batch * npts * 3;

    const int lane = threadIdx.x & 31;
    const int wave = threadIdx.x >> 5;
    const int lo   = lane & 15;   // column / row-within-tile index
    const int hi   = lane >> 4;   // K-half for A/B, row-half for C/D

    const int rowBase = blockIdx.x * 256 + wave * 32;

    // Load this lane's two source points (rows rowBase+lo and rowBase+16+lo).
    const float* sp0 = S + (size_t)(rowBase + lo) * 3;
    const float s0x = sp0[0], s0y = sp0[1], s0z = sp0[2];
    const float x2a = s0x * s0x + s0y * s0y + s0z * s0z;

    const float* sp1 = S + (size_t)(rowBase + 16 + lo) * 3;
    const float s1x = sp1[0], s1y = sp1[1], s1z = sp1[2];
    const float x2b = s1x * s1x + s1y * s1y + s1z * s1z;

    // A tiles hold -2 * source coords, with K=3 slot = 1.0 so that the
    // B tile's K=3 row (|y|^2) is added by the WMMA itself:
    //   D[m][n] = -2*dot(x_m, y_n) + |y_n|^2   (C = 0)
    // 32-bit A 16x4 layout: lanes 0-15: VGPR0=K0, VGPR1=K1;
    //                       lanes 16-31: VGPR0=K2, VGPR1=K3.
    v2f a0, a1;
    a0.x = hi ? (-2.0f * s0z) : (-2.0f * s0x);
    a0.y = hi ? 1.0f          : (-2.0f * s0y);
    a1.x = hi ? (-2.0f * s1z) : (-2.0f * s1x);
    a1.y = hi ? 1.0f          : (-2.0f * s1y);

    const v8f czero = {};

    v8f rowmin0, rowmin1;
#pragma unroll
    for (int v = 0; v < 8; ++v) { rowmin0[v] = 3.4e38f; rowmin1[v] = 3.4e38f; }

    for (int j0 = 0; j0 < npts; j0 += 16) {
        // Every lane loads target point n = j0 + lo (halves redundantly -> cache hit).
        const float* tp = T + (size_t)(j0 + lo) * 3;
        const float tx = tp[0], ty = tp[1], tz = tp[2];
        const float y2 = tx * tx + ty * ty + tz * tz;

        // B 4x16 layout: lanes 0-15: K0(x),K1(y); lanes 16-31: K2(z),K3(|y|^2).
        v2f b;
        b.x = hi ? tz : tx;
        b.y = hi ? y2 : ty;

        v8f d0 = __builtin_amdgcn_wmma_f32_16x16x4_f32(
            false, a0, false, b, (short)0, czero, false, false);
        v8f d1 = __builtin_amdgcn_wmma_f32_16x16x4_f32(
            false, a1, false, b, (short)0, czero, false, false);

#pragma unroll
        for (int v = 0; v < 8; ++v) {
            rowmin0[v] = fminf(rowmin0[v], d0[v]);
            rowmin1[v] = fminf(rowmin1[v], d1[v]);
        }
    }

    // Min-reduce over the 16 columns: xor-shuffle within each half-wave.
#pragma unroll
    for (int m = 8; m >= 1; m >>= 1) {
#pragma unroll
        for (int v = 0; v < 8; ++v) {
            rowmin0[v] = fminf(rowmin0[v], __shfl_xor(rowmin0[v], m, 32));
            rowmin1[v] = fminf(rowmin1[v], __shfl_xor(rowmin1[v], m, 32));
        }
    }

    // rowmin0[v]: min_n (|y|^2 - 2xy) for row m = v + 8*hi        (tile 0)
    // rowmin1[v]: same for row m = 16 + v + 8*hi                  (tile 1)
    // Add |x_m|^2, sqrt, clamp, sum the 16 rows this half-wave owns.
    float local = 0.0f;
#pragma unroll
    for (int v = 0; v < 8; ++v) {
        float x2r0 = __shfl(x2a, v + 8 * hi, 32);
        float x2r1 = __shfl(x2b, v + 8 * hi, 32);
        float dd0 = sqrtf(fmaxf(rowmin0[v] + x2r0, 0.0f));
        float dd1 = sqrtf(fmaxf(rowmin1[v] + x2r1, 0.0f));
        local += fminf(dd0, CHAMFER_THRESH) + fminf(dd1, CHAMFER_THRESH);
    }
    if (lo == 0) atomicAdd(&acc[batch], local);
}

__global__ void chamfer_finalize(const float* __restrict__ acc,
                                 float* __restrict__ out, int nb, float inv2n) {
    int i = blockIdx.x * blockDim.x + threadIdx.x;
    if (i < nb) out[i] = acc[i] * inv2n;
}

extern "C" void kernel_launch(void* const* d_in, const int* in_sizes, int n_in,
                              void* d_out, int out_size, void* d_ws, size_t ws_size,
                              hipStream_t stream) {
    const float* src = (const float*)d_in[0];
    const float* tgt = (const float*)d_in[1];
    float* out = (float*)d_out;
    float* acc = (float*)d_ws;

    const int B = out_size;                      // 2
    const int npts = in_sizes[0] / (B * 3);      // 16384

    chamfer_zero_acc<<<1, 64, 0, stream>>>(acc, B);

    dim3 grid(npts / 256, B, 2);
    chamfer_dir_kernel<<<grid, 256, 0, stream>>>(src, tgt, acc, npts);

    chamfer_finalize<<<1, 64, 0, stream>>>(acc, out, B, 1.0f / (2.0f * (float)npts));
}